// Perplexity_32641751450044
// MI455X (gfx1250) — compile-verified
//
#include <hip/hip_runtime.h>

#define N_BINS 1024
#define HIST_BLOCK 256
#define HIST_WAVES (HIST_BLOCK / 32)
#define HIST_GRID 512
#define EPS_F 1e-8f

typedef __attribute__((ext_vector_type(2))) float v2f;
typedef __attribute__((ext_vector_type(8))) float v8f;

// ---------------------------------------------------------------------------
// Kernel A: zero the 1024 global histogram bins in d_ws (harness poisons ws).
// ---------------------------------------------------------------------------
__global__ void Perplexity_zero_bins(unsigned int* __restrict__ bins) {
    int i = blockIdx.x * blockDim.x + threadIdx.x;
    if (i < N_BINS) bins[i] = 0u;
}

// ---------------------------------------------------------------------------
// Kernel B: streaming histogram.
//   - int4 (b128) coalesced grid-stride loads
//   - per-wave LDS sub-histograms -> ds_add_u32, no inter-wave contention
//   - global_prefetch_b8 one grid-stride ahead
//   - single per-block merge into global bins (L2-resident atomics)
// ---------------------------------------------------------------------------
__global__ void __launch_bounds__(HIST_BLOCK)
Perplexity_hist(const int* __restrict__ idx, unsigned int* __restrict__ gbins,
                int n4, int rem) {
    __shared__ unsigned int lh[HIST_WAVES][N_BINS];

    const int tid  = threadIdx.x;
    const int wave = tid >> 5;

    // zero LDS sub-histograms
    for (int i = tid; i < HIST_WAVES * N_BINS; i += HIST_BLOCK)
        ((unsigned int*)lh)[i] = 0u;
    __syncthreads();

    const int4* __restrict__ v = (const int4*)idx;
    const int stride = gridDim.x * HIST_BLOCK;
    unsigned int* __restrict__ myh = lh[wave];

#pragma unroll 4
    for (int i = blockIdx.x * HIST_BLOCK + tid; i < n4; i += stride) {
        int pf = i + stride;
        if (pf < n4) __builtin_prefetch(&v[pf], 0, 0);   // global_prefetch_b8
        int4 a = v[i];                                   // global_load_b128
        atomicAdd(&myh[a.x & (N_BINS - 1)], 1u);         // ds_add_u32
        atomicAdd(&myh[a.y & (N_BINS - 1)], 1u);
        atomicAdd(&myh[a.z & (N_BINS - 1)], 1u);
        atomicAdd(&myh[a.w & (N_BINS - 1)], 1u);
    }

    // tail (n not multiple of 4): handled by block 0, wave 0
    if (blockIdx.x == 0 && tid < rem) {
        atomicAdd(&lh[0][idx[n4 * 4 + tid] & (N_BINS - 1)], 1u);
    }
    __syncthreads();

    // merge sub-histograms, one global atomic per (block, bin)
    for (int b = tid; b < N_BINS; b += HIST_BLOCK) {
        unsigned int s = 0u;
#pragma unroll
        for (int w = 0; w < HIST_WAVES; ++w) s += lh[w][b];
        if (s) atomicAdd(&gbins[b], s);
    }
}

// ---------------------------------------------------------------------------
// Kernel C: perplexity from bins. One wave (32 lanes, EXEC all-ones).
//   Cross-lane sum done through V_WMMA_F32_16X16X4_F32 (exact f32 path):
//     A slots (2 VGPRs x 32 lanes = 64 elems) all filled with the lane
//     partial -> every lane value appears exactly twice in A, so
//     sum(A) = 2*S regardless of the A layout mapping.
//     B = ones  =>  D[m][n] = rowsum_m(A)  =>  sum(D) = 16*sum(A) = 32*S.
//   A layout-agnostic shfl_xor tree then gives 32*S in every lane.
// ---------------------------------------------------------------------------
__global__ void __launch_bounds__(32)
Perplexity_finish(const unsigned int* __restrict__ bins,
                  float* __restrict__ out, float invN) {
    const int lane = threadIdx.x;   // 0..31, one full wave

    float local = 0.0f;
#pragma unroll
    for (int b = lane; b < N_BINS; b += 32) {
        float p = (float)bins[b] * invN;       // invN = 2^-24 -> exact scale
        local += p * __logf(p + EPS_F);
    }

    v2f a;  a.x = local; a.y = local;          // each lane value twice in A
    v2f b1; b1.x = 1.0f; b1.y = 1.0f;          // B = ones
    v8f c = {};
    v8f d = __builtin_amdgcn_wmma_f32_16x16x4_f32(
        /*neg_a=*/false, a, /*neg_b=*/false, b1,
        /*c_mod=*/(short)0, c, /*reuse_a=*/false, /*reuse_b=*/false);

    float q = d[0] + d[1] + d[2] + d[3] + d[4] + d[5] + d[6] + d[7];
#pragma unroll
    for (int off = 16; off > 0; off >>= 1)
        q += __shfl_xor(q, off, 32);
    // q == 32 * S in every lane
    float S = q * (1.0f / 32.0f);

    if (lane == 0) out[0] = __expf(-S);
}

// ---------------------------------------------------------------------------
extern "C" void kernel_launch(void* const* d_in, const int* in_sizes, int n_in,
                              void* d_out, int out_size, void* d_ws, size_t ws_size,
                              hipStream_t stream) {
    const int* idx = (const int*)d_in[0];
    const int n    = in_sizes[0];            // 16 * 1048576 = 16,777,216
    unsigned int* bins = (unsigned int*)d_ws; // 4 KB of scratch

    Perplexity_zero_bins<<<(N_BINS + 255) / 256, 256, 0, stream>>>(bins);

    const int n4  = n >> 2;
    const int rem = n & 3;
    Perplexity_hist<<<HIST_GRID, HIST_BLOCK, 0, stream>>>(idx, bins, n4, rem);

    Perplexity_finish<<<1, 32, 0, stream>>>(bins, (float*)d_out,
                                            1.0f / (float)n);
}